// HorizontalPooling_6837587935356
// MI455X (gfx1250) — compile-verified
//
#include <hip/hip_runtime.h>

// HorizontalPooling == cumulative max over the batch axis of [B,H,W,C] fp32,
// B=64, H=56, W=56, C=256. Pure HBM-streaming scan: ~411 MB traffic ->
// ~17.6 us floor at 23.3 TB/s. No matrix ops apply (no contraction), so the
// MI455X levers are: wave32 blocks, b128 non-temporal loads/stores (each
// element touched exactly once; 411 MB > 192 MB L2), and deep memory-level
// parallelism (4+ b128 loads in flight per wave before any dependent VALU).

typedef float v4f __attribute__((ext_vector_type(4)));

#define CUMMAX_B 64
// H*W*C = 56*56*256 = 802816 floats per batch plane; /4 = 200704 float4 cols.
#define NCOLS4 200704

__device__ __forceinline__ v4f vmax4(v4f a, v4f b) {
    v4f r;
    r.x = fmaxf(a.x, b.x);
    r.y = fmaxf(a.y, b.y);
    r.z = fmaxf(a.z, b.z);
    r.w = fmaxf(a.w, b.w);
    return r;
}

__global__ __launch_bounds__(256) void
HorizontalPooling_cummax_kernel(const float* __restrict__ in,
                                float* __restrict__ out) {
    const int col = blockIdx.x * blockDim.x + threadIdx.x;
    if (col >= NCOLS4) return;

    const v4f* __restrict__ ip = (const v4f*)in;
    v4f* __restrict__ op       = (v4f*)out;

    long off = (long)col;
    const long S = NCOLS4;  // plane stride in float4 units

    // ---- group 0 (planes 0..3): issue all 4 loads, then the max/store chain.
    v4f a0 = __builtin_nontemporal_load(&ip[off]);
    v4f a1 = __builtin_nontemporal_load(&ip[off + S]);
    v4f a2 = __builtin_nontemporal_load(&ip[off + 2 * S]);
    v4f a3 = __builtin_nontemporal_load(&ip[off + 3 * S]);

    v4f m = a0;
    __builtin_nontemporal_store(m, &op[off]);
    m = vmax4(m, a1);
    __builtin_nontemporal_store(m, &op[off + S]);
    m = vmax4(m, a2);
    __builtin_nontemporal_store(m, &op[off + 2 * S]);
    m = vmax4(m, a3);
    __builtin_nontemporal_store(m, &op[off + 3 * S]);
    off += 4 * S;

    // ---- groups 1..15 (planes 4..63): 4 independent loads clause together;
    // only the short v_max chain is serial. Stores use STOREcnt, so they do
    // not gate the next group's loads (split counters on gfx1250). unroll 2
    // lets the scheduler overlap two groups' load clauses (8 loads in flight).
#pragma unroll 2
    for (int g = 1; g < CUMMAX_B / 4; ++g, off += 4 * S) {
        v4f b0 = __builtin_nontemporal_load(&ip[off]);
        v4f b1 = __builtin_nontemporal_load(&ip[off + S]);
        v4f b2 = __builtin_nontemporal_load(&ip[off + 2 * S]);
        v4f b3 = __builtin_nontemporal_load(&ip[off + 3 * S]);

        m = vmax4(m, b0);
        __builtin_nontemporal_store(m, &op[off]);
        m = vmax4(m, b1);
        __builtin_nontemporal_store(m, &op[off + S]);
        m = vmax4(m, b2);
        __builtin_nontemporal_store(m, &op[off + 2 * S]);
        m = vmax4(m, b3);
        __builtin_nontemporal_store(m, &op[off + 3 * S]);
    }
}

extern "C" void kernel_launch(void* const* d_in, const int* in_sizes, int n_in,
                              void* d_out, int out_size, void* d_ws, size_t ws_size,
                              hipStream_t stream) {
    const float* in = (const float*)d_in[0];
    float* out      = (float*)d_out;

    // 200704 float4 columns / 256 threads = 784 blocks (exact, no tail).
    dim3 block(256);
    dim3 grid((NCOLS4 + block.x - 1) / block.x);
    HorizontalPooling_cummax_kernel<<<grid, block, 0, stream>>>(in, out);
}